// embarkrMPUF_79121887527143
// MI455X (gfx1250) — compile-verified
//
#include <hip/hip_runtime.h>
#include <hip/hip_bf16.h>

typedef __attribute__((ext_vector_type(2))) float v2f;
typedef __attribute__((ext_vector_type(8))) float v8f;

#define D_DIM        65      // PUF_length + 1
#define ROWS_PER_BLK 64      // 4 waves x 16 rows
#define SD_STRIDE    132     // 128 cols + 4 pad (132 % 64 == 4 -> conflict-free halves)

// Fused kernel:
//  Phase 1: per-wave [16 x 65] x [65 x 128] f32 GEMM via v_wmma_f32_16x16x4_f32
//           cols 0..63  = Douts (W_D rows),  cols 64..126 = Souts (W_S rows), col 127 = pad
//           + bias + sigmoid, staged to LDS.
//  Phase 2: closed-form Kronecker cascade:
//           M[j] = prod_i ( bit_i(j) ? S[l_i + (j>>(i+1))] : 1 - S[l_i + (j>>(i+1))] )
//           out[row] = 0.99999 * sum_j Douts[j] * M[j]
__global__ __launch_bounds__(128) void mpuf_fused_kernel(
    const float* __restrict__ phi,
    const float* __restrict__ W_S, const float* __restrict__ b_S,
    const float* __restrict__ W_D, const float* __restrict__ b_D,
    float* __restrict__ out)
{
    __shared__ float SD[ROWS_PER_BLK * SD_STRIDE];

    const int tid   = threadIdx.x;
    const int wave  = tid >> 5;
    const int lane  = tid & 31;
    const int nl    = lane & 15;   // M index (A) / N index (B,C,D) within tile
    const int khalf = lane >> 4;   // hi half-wave takes K = {2,3} of each k-step

    const int rowBlock = blockIdx.x * ROWS_PER_BLK;
    const int rowWave  = rowBlock + wave * 16;

    // ---------------- Phase 1: WMMA GEMM ----------------
    const v8f vzero = {0.f,0.f,0.f,0.f,0.f,0.f,0.f,0.f};
    v8f acc[8];
    #pragma unroll
    for (int t = 0; t < 8; ++t) acc[t] = vzero;

    const float* __restrict__ phiRow = phi + (size_t)(rowWave + nl) * D_DIM;

    // Per-tile B base pointers. Tile 7, lane nl==15 would be W_S row 63 (doesn't
    // exist) -> clamp the address to row 62 (always valid) and zero the loaded
    // value with a loop-invariant per-lane mask (lowers to v_cndmask, no EXEC
    // branch in the hot loop).
    const int   s7     = 48 + nl;                  // tile-7 S row: 48..63
    const bool  s7ok   = (s7 < 63);
    const float s7m    = s7ok ? 1.f : 0.f;         // multiplicative mask
    const float* __restrict__ wS7 = W_S + (s7ok ? s7 : 62) * D_DIM;

    // main K loop: k = 0..63, 16 steps of K=4 (k0+1 <= 63 always in range)
    #pragma unroll 4
    for (int ks = 0; ks < 16; ++ks) {
        const int k0 = ks * 4 + khalf * 2;
        v2f a;
        a.x = phiRow[k0];
        a.y = phiRow[k0 + 1];
        #pragma unroll
        for (int t = 0; t < 8; ++t) {
            v2f b;
            if (t < 4) {                       // Douts columns 0..63
                const float* wp = W_D + (t * 16 + nl) * D_DIM;
                b.x = wp[k0];
                b.y = wp[k0 + 1];
            } else if (t < 7) {                // Souts columns 64..111
                const float* wp = W_S + ((t - 4) * 16 + nl) * D_DIM;
                b.x = wp[k0];
                b.y = wp[k0 + 1];
            } else {                           // Souts columns 112..126 (+ pad 127)
                b.x = wS7[k0]     * s7m;
                b.y = wS7[k0 + 1] * s7m;
            }
            acc[t] = __builtin_amdgcn_wmma_f32_16x16x4_f32(
                false, a, false, b, (short)0, acc[t], false, false);
        }
    }

    // K tail: k = 64..67 (only k==64 is real data).
    // Load at the clamped (uniform) address k=64 and select by half-wave.
    {
        const float kv = (khalf == 0) ? 1.f : 0.f;   // hi half covers k=66,67 -> zero
        v2f a;
        a.x = phiRow[64] * kv;
        a.y = 0.f;
        #pragma unroll
        for (int t = 0; t < 8; ++t) {
            v2f b;
            b.y = 0.f;
            if (t < 4) {
                b.x = W_D[(t * 16 + nl) * D_DIM + 64] * kv;
            } else if (t < 7) {
                b.x = W_S[((t - 4) * 16 + nl) * D_DIM + 64] * kv;
            } else {
                b.x = wS7[64] * s7m * kv;
            }
            acc[t] = __builtin_amdgcn_wmma_f32_16x16x4_f32(
                false, a, false, b, (short)0, acc[t], false, false);
        }
    }

    // bias + sigmoid, stage to LDS.
    // C/D layout: lane nl holds column t*16+nl; VGPR r holds row r + 8*khalf.
    #pragma unroll
    for (int t = 0; t < 8; ++t) {
        const int col = t * 16 + nl;
        float bias;
        if (col < 64)        bias = b_D[col];
        else if (col < 127)  bias = b_S[col - 64];
        else                 bias = 0.f;
        #pragma unroll
        for (int r = 0; r < 8; ++r) {
            const int rloc = wave * 16 + khalf * 8 + r;
            const float v  = acc[t][r] + bias;
            const float sg = 1.0f / (1.0f + __expf(-v));
            SD[rloc * SD_STRIDE + col] = sg;
        }
    }
    __syncthreads();

    // ---------------- Phase 2: cascade + reduce ----------------
    // 2 threads per row, 32 leaves each.
    const int row  = tid >> 1;     // 0..63
    const int half = tid & 1;
    const float* __restrict__ sd = SD + row * SD_STRIDE;

    constexpr int loff[6] = {0, 32, 48, 56, 60, 62};

    float partial = 0.f;
    #pragma unroll
    for (int jj = 0; jj < 32; ++jj) {
        const int j = half * 32 + jj;
        float m = sd[j];                           // Douts[j]
        #pragma unroll
        for (int i = 0; i < 6; ++i) {
            const int idx = loff[i] + (j >> (i + 1));
            const float s = sd[64 + idx];          // Souts[idx]
            m *= ((j >> i) & 1) ? s : (1.0f - s);
        }
        partial += m;
    }
    partial += __shfl_xor(partial, 1, 32);
    if (half == 0) {
        out[rowBlock + row] = partial * 0.99999f;
    }
}

extern "C" void kernel_launch(void* const* d_in, const int* in_sizes, int n_in,
                              void* d_out, int out_size, void* d_ws, size_t ws_size,
                              hipStream_t stream) {
    const float* phi = (const float*)d_in[0];
    const float* W_S = (const float*)d_in[1];
    const float* b_S = (const float*)d_in[2];
    const float* W_D = (const float*)d_in[3];
    const float* b_D = (const float*)d_in[4];
    // d_in[5] = Snum (scalar, hardcoded as 6 in the kernel structure)
    float* out = (float*)d_out;

    const int B      = in_sizes[0] / D_DIM;   // 524288
    const int blocks = B / ROWS_PER_BLK;      // 8192

    mpuf_fused_kernel<<<blocks, 128, 0, stream>>>(phi, W_S, b_S, W_D, b_D, out);
}